// PointTrans_Layer_down_86071144612519
// MI455X (gfx1250) — compile-verified
//
#include <hip/hip_runtime.h>

// Problem constants (match reference)
#define NN   262144          // nodes
#define EE   (NN * 16)       // edges
#define CC   64              // channels in/out
#define NVD  20              // voxels per dim
#define NVOX 8000            // total voxels
#define OUTC 67              // C_OUT + 3
#define BN_EPS 1e-5f

typedef __attribute__((ext_vector_type(2))) float v2f;
typedef __attribute__((ext_vector_type(8))) float v8f;

// ---------------------------------------------------------------- zero fill
__global__ void zero_kernel(float* __restrict__ p, int n) {
    int i = blockIdx.x * blockDim.x + threadIdx.x;
    if (i < n) p[i] = 0.0f;
}

// ---------------------------------------------------------------- GEMM via V_WMMA_F32_16X16X4_F32
// h[N,64] = x[N,64] @ W[64,64] + b[64]
// One wave computes a 16-row x 64-col tile: 4 accumulators, 16 K-steps of 4.
__global__ void gemm_wmma_kernel(const float* __restrict__ x,
                                 const float* __restrict__ W,
                                 const float* __restrict__ b,
                                 float* __restrict__ h) {
    const int lane = threadIdx.x & 31;
    const int wave = threadIdx.x >> 5;
    const int tile = blockIdx.x * 4 + wave;   // 16-row tile index
    const int m0   = tile * 16;
    const int half = lane >> 4;               // 0: lanes 0-15, 1: lanes 16-31
    const int l16  = lane & 15;

    v8f acc[4];
    acc[0] = (v8f){}; acc[1] = (v8f){}; acc[2] = (v8f){}; acc[3] = (v8f){};

    // A fragment source row for this lane (lane = M within the 16-row tile)
    const float* __restrict__ xr = x + (size_t)(m0 + l16) * CC;

    #pragma unroll
    for (int kk = 0; kk < 16; ++kk) {
        const int k  = kk * 4;
        const int ka = k + 2 * half;          // half-wave splits K: {k,k+1} / {k+2,k+3}
        v2f a;
        a.x = xr[ka];
        a.y = xr[ka + 1];
        #pragma unroll
        for (int nt = 0; nt < 4; ++nt) {
            const int n = nt * 16 + l16;      // lane = N within the 16-col tile
            v2f bb;
            bb.x = W[(size_t)ka * CC + n];
            bb.y = W[(size_t)(ka + 1) * CC + n];
            // 8 args: (neg_a, A, neg_b, B, c_mod, C, reuse_a, reuse_b)
            acc[nt] = __builtin_amdgcn_wmma_f32_16x16x4_f32(
                false, a, false, bb, (short)0, acc[nt], false, false);
        }
    }

    // C/D layout: VGPR r holds M = m0 + r + 8*half, N = n0 + l16
    #pragma unroll
    for (int nt = 0; nt < 4; ++nt) {
        const int n = nt * 16 + l16;
        const float bias = b[n];
        #pragma unroll
        for (int r = 0; r < 8; ++r) {
            const int m = m0 + r + 8 * half;
            h[(size_t)m * CC + n] = acc[nt][r] + bias;
        }
    }
}

// ---------------------------------------------------------------- per-channel sum / sumsq
// 512 blocks x 256 threads; each block covers 512 rows; thread owns channel tid&63.
__global__ void stats_kernel(const float* __restrict__ h,
                             float* __restrict__ sum,
                             float* __restrict__ sq) {
    const int c   = threadIdx.x & 63;
    const int sub = threadIdx.x >> 6;             // 0..3
    const int r0  = blockIdx.x * 512 + sub;
    float s = 0.0f, q = 0.0f;
    #pragma unroll 4
    for (int i = 0; i < 128; ++i) {
        float v = h[(size_t)(r0 + i * 4) * CC + c];
        s += v;
        q += v * v;
    }
    atomicAdd(&sum[c], s);
    atomicAdd(&sq[c], q);
}

// ---------------------------------------------------------------- BN scale/shift
__global__ void bnstat_kernel(const float* __restrict__ sum,
                              const float* __restrict__ sq,
                              const float* __restrict__ gamma,
                              const float* __restrict__ beta,
                              float* __restrict__ scale,
                              float* __restrict__ shift) {
    const int c = threadIdx.x;
    const float inv_n = 1.0f / (float)NN;
    float mean = sum[c] * inv_n;
    float var  = sq[c] * inv_n - mean * mean;     // biased variance
    float sc   = gamma[c] * rsqrtf(var + BN_EPS);
    scale[c] = sc;
    shift[c] = beta[c] - mean * sc;
}

// ---------------------------------------------------------------- BN + ReLU; seed h2 with self-loop
__global__ void bnrelu_kernel(float* __restrict__ h,
                              float* __restrict__ h2,
                              const float* __restrict__ scale,
                              const float* __restrict__ shift) {
    const int gid = blockIdx.x * blockDim.x + threadIdx.x;   // < N*64 = 16.7M
    const int c = gid & 63;
    float v = fmaxf(h[gid] * scale[c] + shift[c], 0.0f);
    h[gid]  = v;   // scatter-max source (frozen)
    h2[gid] = v;   // scatter-max destination (self-loop seed)
}

// ---------------------------------------------------------------- edge scatter-max
// 64 consecutive threads per edge (coalesced channels). Values >= 0 after ReLU,
// so int-bitwise atomicMax is order-correct; skip atomic when source is 0.
__global__ void edgemax_kernel(const int* __restrict__ row,
                               const int* __restrict__ col,
                               const float* __restrict__ h,
                               int* __restrict__ h2i) {
    const unsigned gid = blockIdx.x * blockDim.x + threadIdx.x;  // < E*64 = 268M
    const int c = gid & 63;
    const int e = gid >> 6;
    const int r  = row[e];
    const int cl = col[e];
    float v = h[(size_t)r * CC + c];
    if (v > 0.0f)
        atomicMax(&h2i[(size_t)cl * CC + c], __float_as_int(v));
}

// ---------------------------------------------------------------- voxel scatter (sum)
__global__ void voxscatter_kernel(const float* __restrict__ h2,
                                  const float* __restrict__ pos,
                                  float* __restrict__ out,   // [NVOX, 67], accumulates sums
                                  float* __restrict__ cnt) { // [NVOX]
    const int gid = blockIdx.x * blockDim.x + threadIdx.x;   // < N*64
    const int c = gid & 63;
    const int n = gid >> 6;
    const float px = pos[(size_t)n * 3 + 0];
    const float py = pos[(size_t)n * 3 + 1];
    const float pz = pos[(size_t)n * 3 + 2];
    // GRID = 0.5 -> pos/GRID == pos*2 exactly
    int vx = min(max((int)floorf(px * 2.0f), 0), NVD - 1);
    int vy = min(max((int)floorf(py * 2.0f), 0), NVD - 1);
    int vz = min(max((int)floorf(pz * 2.0f), 0), NVD - 1);
    const int vox = (vx * NVD + vy) * NVD + vz;

    atomicAdd(&out[(size_t)vox * OUTC + c], h2[gid]);
    if (c == 0) atomicAdd(&cnt[vox], 1.0f);
    if (c < 3) {
        float pv = pos[(size_t)n * 3 + c];
        atomicAdd(&out[(size_t)vox * OUTC + CC + c], pv);
    }
}

// ---------------------------------------------------------------- divide by count
__global__ void voxfinal_kernel(float* __restrict__ out,
                                const float* __restrict__ cnt) {
    const int gid = blockIdx.x * blockDim.x + threadIdx.x;
    if (gid >= NVOX * OUTC) return;
    const int v = gid / OUTC;
    out[gid] = out[gid] / fmaxf(cnt[v], 1.0f);
}

// ================================================================ launch
extern "C" void kernel_launch(void* const* d_in, const int* in_sizes, int n_in,
                              void* d_out, int out_size, void* d_ws, size_t ws_size,
                              hipStream_t stream) {
    const float* x     = (const float*)d_in[0];   // [N, 64]
    const float* pos   = (const float*)d_in[1];   // [N, 3]
    const int*   eidx  = (const int*)  d_in[2];   // [2, E]
    const float* W     = (const float*)d_in[3];   // [64, 64]
    const float* b     = (const float*)d_in[4];   // [64]
    const float* gamma = (const float*)d_in[5];   // [64]
    const float* beta  = (const float*)d_in[6];   // [64]
    float* out = (float*)d_out;                   // [NVOX, 67]

    const int* row = eidx;        // edge_index[0]
    const int* col = eidx + EE;   // edge_index[1]

    // workspace layout
    float* ws    = (float*)d_ws;
    float* h     = ws;                            // N*64
    float* h2    = h  + (size_t)NN * CC;          // N*64
    float* sum   = h2 + (size_t)NN * CC;          // 64
    float* sq    = sum + 64;                      // 64
    float* scale = sq + 64;                       // 64
    float* shift = scale + 64;                    // 64
    float* cnt   = shift + 64;                    // NVOX

    // 1) zero accumulators (sum/sq/scale/shift/cnt contiguous) and output
    {
        const int n_acc = 64 * 4 + NVOX;          // 8256
        zero_kernel<<<(n_acc + 255) / 256, 256, 0, stream>>>(sum, n_acc);
        const int n_out = NVOX * OUTC;            // 536000
        zero_kernel<<<(n_out + 255) / 256, 256, 0, stream>>>(out, n_out);
    }

    // 2) h = x @ W + b (WMMA fp32). 16384 row-tiles, 4 waves/block -> 4096 blocks.
    gemm_wmma_kernel<<<NN / 16 / 4, 128, 0, stream>>>(x, W, b, h);

    // 3) BN batch statistics
    stats_kernel<<<512, 256, 0, stream>>>(h, sum, sq);
    bnstat_kernel<<<1, 64, 0, stream>>>(sum, sq, gamma, beta, scale, shift);

    // 4) BN + ReLU (in place) and seed h2 with self-loop values
    bnrelu_kernel<<<(NN * CC) / 256, 256, 0, stream>>>(h, h2, scale, shift);

    // 5) scatter-max over edges
    edgemax_kernel<<<(unsigned)(((size_t)EE * CC) / 256), 256, 0, stream>>>(
        row, col, h, (int*)h2);

    // 6) voxel-grid sums + counts
    voxscatter_kernel<<<(NN * CC) / 256, 256, 0, stream>>>(h2, pos, out, cnt);

    // 7) mean = sum / max(cnt, 1)
    voxfinal_kernel<<<(NVOX * OUTC + 255) / 256, 256, 0, stream>>>(out, cnt);
}